// TransR_85074712199923
// MI455X (gfx1250) — compile-verified
//
#include <hip/hip_runtime.h>

// ---------------------------------------------------------------------------
// TransR scoring on MI455X (gfx1250), f32 end-to-end.
//   * one block per (pos,neg) pair; nr==pr so the 256KB projection matrix is
//     loaded ONCE and feeds 4 vec-mats (h+,t+,h-,t-) via V_WMMA_F32_16X16X4_F32
//   * proj double-buffered through LDS by the Tensor Data Mover (one
//     TENSOR_LOAD_TO_LDS per 32-row chunk, hardware row padding to PSTR=272,
//     s_wait_tensorcnt); falls back to GLOBAL_LOAD_ASYNC_TO_LDS_B128, then to
//     synchronous staging
//   * A operand pre-swizzled into LDS -> one uniform ds_load_b64 per K-step
//   * HBM-bound (~1GB proj + 102MB reg pass @ 23.3TB/s); f32 WMMA keeps
//     reference precision (inputs are f32 in HBM, low precision saves 0 bytes)
// ---------------------------------------------------------------------------

typedef __attribute__((ext_vector_type(2))) float v2f;
typedef __attribute__((ext_vector_type(8))) float v8f;
typedef __attribute__((__vector_size__(4 * sizeof(int)))) int v4i;

#define ENT_NUM   100000
#define REL_NUM   1000
#define DIM       256
#define BATCH_N   4096
#define MARGIN_V  4.0f
#define ALPHA_V   0.01f

#define KC     32    // projection rows per LDS chunk
#define NCHUNK (DIM / KC)             // 8
#define PSTR   272   // LDS row stride (floats); 272 % 64 == 16 -> wave halves on disjoint banks
#define NSTEP  (DIM / 4)              // 64 WMMA K-steps

#if __has_builtin(__builtin_amdgcn_tensor_load_to_lds) && \
    __has_builtin(__builtin_amdgcn_s_wait_tensorcnt)
#define HAVE_TDM 1
typedef __attribute__((__vector_size__(16))) unsigned int v4u;
typedef __attribute__((__vector_size__(32))) int          v8i;
#else
#define HAVE_TDM 0
#endif

#if __has_builtin(__builtin_amdgcn_global_load_async_to_lds_b128) && \
    __has_builtin(__builtin_amdgcn_s_wait_asynccnt)
#define HAVE_ASYNC_LDS 1
#else
#define HAVE_ASYNC_LDS 0
#endif

typedef __attribute__((address_space(1))) v4i gbl_v4i;   // async global src type
typedef __attribute__((address_space(3))) v4i lds_v4i;   // async LDS dst type

__device__ __forceinline__ unsigned lds_addr_of(const void* p) {
  // AS3 pointer value == byte offset within the workgroup's LDS allocation
  return (unsigned)(size_t)(__attribute__((address_space(3))) const void*)p;
}

__global__ __launch_bounds__(256) void transr_pair_kernel(
    const float* __restrict__ ent_emb,
    const float* __restrict__ rel_emb,
    const float* __restrict__ transfer,
    const int*   __restrict__ pos,
    const int*   __restrict__ neg,
    float*       __restrict__ margin_sum)
{
  __shared__ float vecs[4 * DIM];            // raw entity rows h+, t+, h-, t-
  __shared__ float apack[NSTEP * 64];        // pre-swizzled A fragments (16KB)
  __shared__ float ptile[2][KC * PSTR];      // double-buffered proj chunk (~68KB)
  __shared__ float red[2];                   // pos/neg L1 partial sums

  const int pair = blockIdx.x;
  const int tid  = threadIdx.x;
  const int lane = tid & 31;                 // wave32
  const int wave = tid >> 5;                 // 0..7

  const int ph = pos[pair*3 + 0];
  const int pr = pos[pair*3 + 1];            // neg relation == pos relation
  const int pt = pos[pair*3 + 2];
  const int nh = neg[pair*3 + 0];
  const int nt = neg[pair*3 + 2];

  // ---- stage the four entity vectors (coalesced) --------------------------
  for (int i = tid; i < 4*DIM; i += 256) {
    const int mm = i >> 8;
    const int k  = i & 255;
    const int e  = (mm == 0) ? ph : (mm == 1) ? pt : (mm == 2) ? nh : nt;
    vecs[i] = ent_emb[(size_t)e * DIM + k];
  }
  if (tid < 2) red[tid] = 0.0f;
  __syncthreads();

  // ---- pre-swizzle A fragments: one ds_load_b64 per K-step in the hot loop.
  // 16x4 f32 A layout: lane L holds row m=L&15, K = 4s + 2*(L>>4) + {0,1}.
  // Rows 4..15 are zero so the C tile's unused rows stay exactly 0.
  for (int i = tid; i < NSTEP*64; i += 256) {
    const int s  = i >> 6;
    const int l  = (i >> 1) & 31;
    const int h  = i & 1;
    const int mm = l & 15;
    const int k  = 4*s + ((l >> 4) << 1) + h;
    apack[i] = (mm < 4) ? vecs[mm*DIM + k] : 0.0f;
  }
  // visibility of apack is covered by the first in-loop barrier below

  const float* proj = transfer + (size_t)pr * (DIM * DIM);

  v8f acc0 = {};                             // C tile, columns [n0, n0+16)
  v8f acc1 = {};                             // C tile, columns [n1, n1+16)

  const int n0   = wave * 32;
  const int n1   = n0 + 16;
  // 4x16 f32 B layout: VGPR0 rows K0/K1 across wave halves, VGPR1 rows K2/K3.
  const int bn   = lane & 15;
  const int brow = lane >> 4;                // 0 or 1

#if HAVE_TDM
  // ---- Tensor Data Mover: one TENSOR_LOAD_TO_LDS per 32x256 f32 chunk. ----
  // D# group1: data_size=2(4B); pad_enable=1, pad_interval=7(256 dwords),
  // pad_amount=15(16 dwords) -> rows land at 272-float stride == PSTR.
  // tensor/tile: dim0=256 (row), dim1=32 rows, stride0=256.
  const int prs = __builtin_amdgcn_readfirstlane(pr);
  const unsigned long long gbase =
      (unsigned long long)(size_t)(transfer + (size_t)prs * (DIM * DIM));
  const unsigned ldsA = lds_addr_of(&ptile[0][0]);
  const unsigned ldsB = lds_addr_of(&ptile[1][0]);

  v8i g1;
  g1[0] = (int)((2u << 16) | (1u << 20) | (7u << 22) | (15u << 25)); // 0x1FD20000
  g1[1] = (int)(256u << 16);   // tensor_dim0 = 256 (bits 79:48, low half)
  g1[2] = (int)(32u  << 16);   // tensor_dim1 = 32  (bits 111:80, low half)
  g1[3] = (int)(256u << 16);   // tile_dim0   = 256 (bits 127:112)
  g1[4] = 32;                  // tile_dim1   = 32  (bits 143:128)
  g1[5] = 256;                 // tensor_dim0_stride = 256 (bits 207:160)
  g1[6] = 0;
  g1[7] = 0;
  const v4i gz  = {0, 0, 0, 0};                  // groups 2/3 unused (<=2D tensor)
  const v8i gz8 = {0, 0, 0, 0, 0, 0, 0, 0};      // 5th descriptor word group

#define TDM_ISSUE(c, buf)                                                       \
  {                                                                             \
    const unsigned long long ga =                                               \
        gbase + (unsigned long long)(c) * (KC * DIM * 4);                       \
    v4u g0;                                                                     \
    g0[0] = 1u;                              /* count=1, user mode */           \
    g0[1] = (buf) ? ldsB : ldsA;             /* lds_addr */                     \
    g0[2] = (unsigned)ga;                    /* global_addr[31:0] */            \
    g0[3] = (unsigned)((ga >> 32) & 0x1FFFFFFu) | 0x80000000u; /* +type=2 */    \
    __builtin_amdgcn_tensor_load_to_lds(g0, g1, gz, gz, gz8, 0);                \
  }

  if (wave == 0) TDM_ISSUE(0, 0)
#pragma unroll
  for (int c = 0; c < NCHUNK; ++c) {
    if (wave == 0) {
      if (c + 1 < NCHUNK) {
        TDM_ISSUE(c + 1, (c + 1) & 1)        // prefetch next chunk
        // TDM ops of one wave complete in order: <=1 pending => chunk c landed
        __builtin_amdgcn_s_wait_tensorcnt(1);
      } else {
        __builtin_amdgcn_s_wait_tensorcnt(0);
      }
    }
    __syncthreads();                         // releases only after wave0 waited
    const float* pb = ptile[c & 1];
    const float* ap = apack + c * (8 * 64);
#pragma unroll
    for (int kc = 0; kc < KC/4; ++kc) {
      const int kk = kc * 4;
      v2f a;
      a.x = ap[kc*64 + lane*2 + 0];
      a.y = ap[kc*64 + lane*2 + 1];
      v2f b0, b1;
      b0.x = pb[(kk + brow)*PSTR     + n0 + bn];
      b0.y = pb[(kk + 2 + brow)*PSTR + n0 + bn];
      b1.x = pb[(kk + brow)*PSTR     + n1 + bn];
      b1.y = pb[(kk + 2 + brow)*PSTR + n1 + bn];
      acc0 = __builtin_amdgcn_wmma_f32_16x16x4_f32(
                 false, a, false, b0, (short)0, acc0, false, false);
      acc1 = __builtin_amdgcn_wmma_f32_16x16x4_f32(
                 false, a, false, b1, (short)0, acc1, false, false);
    }
    __syncthreads();   // everyone done with ptile[c&1] before it is refilled
  }
#undef TDM_ISSUE
#elif HAVE_ASYNC_LDS
  // ---- per-lane async copies: 8 x b128 per thread per chunk ---------------
#define ISSUE_CHUNK(c, buf)                                                     \
  {                                                                             \
    _Pragma("unroll")                                                           \
    for (int j = 0; j < 8; ++j) {                                               \
      const int q   = tid + j*256;                                              \
      const int row = q >> 6;                                                   \
      const int col = (q & 63) << 2;                                            \
      __builtin_amdgcn_global_load_async_to_lds_b128(                           \
          (gbl_v4i*)(proj + (size_t)(c)*(KC*DIM) + (size_t)q*4),                \
          (lds_v4i*)&ptile[(buf)][row*PSTR + col], 0, 0);                       \
    }                                                                           \
  }

  ISSUE_CHUNK(0, 0)
#pragma unroll
  for (int c = 0; c < NCHUNK; ++c) {
    if (c + 1 < NCHUNK) {
      ISSUE_CHUNK(c + 1, (c + 1) & 1)
      __builtin_amdgcn_s_wait_asynccnt(8);
    } else {
      __builtin_amdgcn_s_wait_asynccnt(0);
    }
    __syncthreads();
    const float* pb = ptile[c & 1];
    const float* ap = apack + c * (8 * 64);
#pragma unroll
    for (int kc = 0; kc < KC/4; ++kc) {
      const int kk = kc * 4;
      v2f a;
      a.x = ap[kc*64 + lane*2 + 0];
      a.y = ap[kc*64 + lane*2 + 1];
      v2f b0, b1;
      b0.x = pb[(kk + brow)*PSTR     + n0 + bn];
      b0.y = pb[(kk + 2 + brow)*PSTR + n0 + bn];
      b1.x = pb[(kk + brow)*PSTR     + n1 + bn];
      b1.y = pb[(kk + 2 + brow)*PSTR + n1 + bn];
      acc0 = __builtin_amdgcn_wmma_f32_16x16x4_f32(
                 false, a, false, b0, (short)0, acc0, false, false);
      acc1 = __builtin_amdgcn_wmma_f32_16x16x4_f32(
                 false, a, false, b1, (short)0, acc1, false, false);
    }
    __syncthreads();
  }
#undef ISSUE_CHUNK
#else
  // ---- synchronous global->reg->LDS staging (single buffer) ---------------
  const float4* proj4 = reinterpret_cast<const float4*>(proj);
  for (int c = 0; c < NCHUNK; ++c) {
    float4 stage[8];
#pragma unroll
    for (int j = 0; j < 8; ++j) stage[j] = proj4[c*2048 + tid + j*256];
    __syncthreads();
#pragma unroll
    for (int j = 0; j < 8; ++j) {
      const int q   = tid + j*256;
      const int row = q >> 6;
      const int col = (q & 63) << 2;
      float* d = &ptile[0][row*PSTR + col];
      d[0] = stage[j].x; d[1] = stage[j].y; d[2] = stage[j].z; d[3] = stage[j].w;
    }
    __syncthreads();
    const float* pb = ptile[0];
    const float* ap = apack + c * (8 * 64);
#pragma unroll
    for (int kc = 0; kc < KC/4; ++kc) {
      const int kk = kc * 4;
      v2f a;
      a.x = ap[kc*64 + lane*2 + 0];
      a.y = ap[kc*64 + lane*2 + 1];
      v2f b0, b1;
      b0.x = pb[(kk + brow)*PSTR     + n0 + bn];
      b0.y = pb[(kk + 2 + brow)*PSTR + n0 + bn];
      b1.x = pb[(kk + brow)*PSTR     + n1 + bn];
      b1.y = pb[(kk + 2 + brow)*PSTR + n1 + bn];
      acc0 = __builtin_amdgcn_wmma_f32_16x16x4_f32(
                 false, a, false, b0, (short)0, acc0, false, false);
      acc1 = __builtin_amdgcn_wmma_f32_16x16x4_f32(
                 false, a, false, b1, (short)0, acc1, false, false);
    }
  }
#endif

  // C layout: lanes 0..15 / VGPR v hold M=v.  Rows: 0=h+P, 1=t+P, 2=h-P, 3=t-P.
  if (lane < 16) {
    const float rv0 = rel_emb[(size_t)pr*DIM + n0 + bn];
    const float rv1 = rel_emb[(size_t)pr*DIM + n1 + bn];
    const float psc = fabsf(acc0[0] - acc0[1] + rv0) + fabsf(acc1[0] - acc1[1] + rv1);
    const float nsc = fabsf(acc0[2] - acc0[3] + rv0) + fabsf(acc1[2] - acc1[3] + rv1);
    atomicAdd(&red[0], psc);
    atomicAdd(&red[1], nsc);
  }
  __syncthreads();
  if (tid == 0) {
    const float diff = red[0] - red[1] + MARGIN_V;   // pos_d - neg_d + margin
    if (diff > 0.0f) atomicAdd(margin_sum, diff);
  }
}

// sum over rows of relu(||row||_2 - 1); one wave32 per row
__global__ __launch_bounds__(256) void norm_penalty_kernel(
    const float* __restrict__ emb, int rows, float* __restrict__ accum)
{
  __shared__ float bsum;
  if (threadIdx.x == 0) bsum = 0.0f;
  __syncthreads();

  const int lane = threadIdx.x & 31;
  const int row  = blockIdx.x * 8 + (threadIdx.x >> 5);
  float pen = 0.0f;
  if (row < rows) {
    const float* p = emb + (size_t)row * DIM;
    float s = 0.0f;
#pragma unroll
    for (int k = 0; k < DIM/32; ++k) {
      const float v = p[lane + k*32];
      s += v * v;
    }
#pragma unroll
    for (int off = 16; off; off >>= 1) s += __shfl_xor(s, off, 32);
    pen = sqrtf(s) - 1.0f;
  }
  if (lane == 0 && pen > 0.0f) atomicAdd(&bsum, pen);
  __syncthreads();
  if (threadIdx.x == 0) atomicAdd(accum, bsum);
}

__global__ void init_ws_kernel(float* ws) {
  if (threadIdx.x < 3) ws[threadIdx.x] = 0.0f;
}

__global__ void finalize_kernel(const float* __restrict__ ws, float* __restrict__ out) {
  out[0] = ws[0] * (1.0f / BATCH_N)
         + ALPHA_V * ((ws[1] + ws[2]) * (1.0f / (ENT_NUM + REL_NUM)));
}

extern "C" void kernel_launch(void* const* d_in, const int* in_sizes, int n_in,
                              void* d_out, int out_size, void* d_ws, size_t ws_size,
                              hipStream_t stream) {
  const float* ent_emb  = (const float*)d_in[0];
  const float* rel_emb  = (const float*)d_in[1];
  const float* transfer = (const float*)d_in[2];
  const int*   pos      = (const int*)d_in[3];
  const int*   neg      = (const int*)d_in[4];
  float*       out      = (float*)d_out;
  float*       ws       = (float*)d_ws;   // ws[0]=margin_sum ws[1]=ent_scale ws[2]=rel_scale

  init_ws_kernel<<<1, 32, 0, stream>>>(ws);
  transr_pair_kernel<<<BATCH_N, 256, 0, stream>>>(ent_emb, rel_emb, transfer,
                                                  pos, neg, ws + 0);
  norm_penalty_kernel<<<(ENT_NUM + 7) / 8, 256, 0, stream>>>(ent_emb, ENT_NUM, ws + 1);
  norm_penalty_kernel<<<(REL_NUM + 7) / 8, 256, 0, stream>>>(rel_emb, REL_NUM, ws + 2);
  finalize_kernel<<<1, 1, 0, stream>>>(ws, out);
}